// DirSAGEConv_62723702391594
// MI455X (gfx1250) — compile-verified
//
#include <hip/hip_runtime.h>
#include <hip/hip_bf16.h>

#define D 128
#define ALPHA 0.5f

typedef __attribute__((ext_vector_type(2))) float v2f;
typedef __attribute__((ext_vector_type(8))) float v8f;

// ---------------------------------------------------------------------------
// Zero the accumulation workspace (sums + degree counters).
// ---------------------------------------------------------------------------
__global__ void dirsage_zero_kernel(float* __restrict__ ws, long long n) {
    long long i = (long long)blockIdx.x * blockDim.x + threadIdx.x;
    long long stride = (long long)gridDim.x * blockDim.x;
    for (; i < n; i += stride) ws[i] = 0.0f;
}

// ---------------------------------------------------------------------------
// Edge scatter: one wave32 per edge. Lane L moves floats [4L, 4L+3] of the
// 128-wide feature row. Both directions handled in one pass so the edge
// endpoints are loaded once.
//   sum_s2d[dst] += x[src]; cnt_dst[dst] += 1
//   sum_d2s[src] += x[dst]; cnt_src[src] += 1
// ---------------------------------------------------------------------------
__global__ __launch_bounds__(256) void dirsage_scatter_kernel(
    const float* __restrict__ x,
    const int*   __restrict__ src,
    const int*   __restrict__ dst,
    float* __restrict__ sum_s2d,
    float* __restrict__ sum_d2s,
    float* __restrict__ cnt_dst,
    float* __restrict__ cnt_src,
    int n_edges) {
    long long gid = (long long)blockIdx.x * blockDim.x + threadIdx.x;
    int e    = (int)(gid >> 5);
    int lane = (int)(gid & 31);
    if (e >= n_edges) return;

    int s = src[e];
    int d = dst[e];

    const float4 xs = *(const float4*)(x + (long long)s * D + lane * 4);
    const float4 xd = *(const float4*)(x + (long long)d * D + lane * 4);

    float* ps = sum_s2d + (long long)d * D + lane * 4;   // x[src] -> dst bucket
    float* pd = sum_d2s + (long long)s * D + lane * 4;   // x[dst] -> src bucket

    unsafeAtomicAdd(ps + 0, xs.x);
    unsafeAtomicAdd(ps + 1, xs.y);
    unsafeAtomicAdd(ps + 2, xs.z);
    unsafeAtomicAdd(ps + 3, xs.w);
    unsafeAtomicAdd(pd + 0, xd.x);
    unsafeAtomicAdd(pd + 1, xd.y);
    unsafeAtomicAdd(pd + 2, xd.z);
    unsafeAtomicAdd(pd + 3, xd.w);

    if (lane == 0) {
        unsafeAtomicAdd(cnt_dst + d, 1.0f);
        unsafeAtomicAdd(cnt_src + s, 1.0f);
    }
}

// ---------------------------------------------------------------------------
// Fused triple GEMM with fp32 WMMA (V_WMMA_F32_16X16X4_F32), full precision.
//   out = x @ Ws^T + (1-a) * mean_s2d @ W1^T + a * mean_d2s @ W2^T + bias
// The (1-a)/deg and a/deg scalings are folded into the A fragments (WMMA is
// linear in A), so one v8f accumulator serves all three products.
// Block = 256 threads = 8 waves; block owns 16 nodes (M tile), wave w owns
// output columns [16w, 16w+15] (N tile). K = 128 in steps of 4.
// ---------------------------------------------------------------------------
__global__ __launch_bounds__(256) void dirsage_wmma_gemm_kernel(
    const float* __restrict__ x,
    const float* __restrict__ sum_s2d,
    const float* __restrict__ sum_d2s,
    const float* __restrict__ cnt_dst,
    const float* __restrict__ cnt_src,
    const float* __restrict__ W1, const float* __restrict__ b1,
    const float* __restrict__ W2, const float* __restrict__ b2,
    const float* __restrict__ Ws, const float* __restrict__ bs,
    float* __restrict__ out, int n_nodes) {
    __shared__ float ldsX [16 * D];
    __shared__ float ldsM1[16 * D];
    __shared__ float ldsM2[16 * D];
    __shared__ float s1s[16];
    __shared__ float s2s[16];

    const int tid = threadIdx.x;
    const long long tile0 = (long long)blockIdx.x * 16;

    if (tid < 16) {
        long long node = tile0 + tid;
        if (node >= n_nodes) node = n_nodes - 1;
        s1s[tid] = (1.0f - ALPHA) / fmaxf(cnt_dst[node], 1.0f);
        s2s[tid] = ALPHA / fmaxf(cnt_src[node], 1.0f);
    }
    __syncthreads();

    // Stage (already scaled) A rows for all three products: 3 * 16 * 128 f32.
    for (int idx = tid; idx < 16 * D; idx += 256) {
        int m = idx >> 7;                       // node row in tile (D == 128)
        long long node = tile0 + m;
        if (node >= n_nodes) node = n_nodes - 1;
        long long g = node * D + (idx & (D - 1));
        ldsX [idx] = x[g];
        ldsM1[idx] = sum_s2d[g] * s1s[m];
        ldsM2[idx] = sum_d2s[g] * s2s[m];
    }
    __syncthreads();

    const int wave = tid >> 5;            // N tile id, 0..7
    const int lane = tid & 31;
    const int mrow = lane & 15;           // A row owned by this lane
    const int koff = (lane >> 4) * 2;     // lanes 0-15: K{0,1}; 16-31: K{2,3}
    const int ncol = wave * 16 + (lane & 15);

    v8f acc = {};

    const float* Lbase[3] = { ldsX, ldsM1, ldsM2 };
    const float* Wbase[3] = { Ws,   W1,    W2    };

    for (int mat = 0; mat < 3; ++mat) {
        const float* L = Lbase[mat] + mrow * D + koff;
        const float* W = Wbase[mat] + (long long)ncol * D + koff;
#pragma unroll
        for (int k0 = 0; k0 < D; k0 += 4) {
            v2f a = *(const v2f*)(L + k0);   // A[mrow][k0+koff .. +1]
            v2f b = *(const v2f*)(W + k0);   // B[k][ncol] = W[ncol][k]
            acc = __builtin_amdgcn_wmma_f32_16x16x4_f32(
                /*neg_a=*/false, a, /*neg_b=*/false, b,
                /*c_mod=*/(short)0, acc, /*reuse_a=*/false, /*reuse_b=*/false);
        }
    }

    const float bias = bs[ncol] + (1.0f - ALPHA) * b1[ncol] + ALPHA * b2[ncol];

#pragma unroll
    for (int r = 0; r < 8; ++r) {
        long long row = tile0 + r + (lane >> 4) * 8;   // C layout: VGPR r -> M=r / r+8
        if (row < n_nodes) out[row * D + ncol] = acc[r] + bias;
    }
}

// ---------------------------------------------------------------------------
// Inputs (setup_inputs order): x, edge_index, W1, b1, W2, b2, Ws, bs
// ---------------------------------------------------------------------------
extern "C" void kernel_launch(void* const* d_in, const int* in_sizes, int n_in,
                              void* d_out, int out_size, void* d_ws, size_t ws_size,
                              hipStream_t stream) {
    const float* x  = (const float*)d_in[0];
    const int*   ei = (const int*)  d_in[1];
    const float* W1 = (const float*)d_in[2];
    const float* b1 = (const float*)d_in[3];
    const float* W2 = (const float*)d_in[4];
    const float* b2 = (const float*)d_in[5];
    const float* Ws = (const float*)d_in[6];
    const float* bs = (const float*)d_in[7];
    float* out = (float*)d_out;

    const int n_nodes = in_sizes[0] / D;
    const int n_edges = in_sizes[1] / 2;

    float* ws       = (float*)d_ws;
    float* sum_s2d  = ws;
    float* sum_d2s  = ws + (long long)n_nodes * D;
    float* cnt_dst  = ws + 2LL * n_nodes * D;
    float* cnt_src  = cnt_dst + n_nodes;

    const long long zero_n = 2LL * n_nodes * D + 2LL * n_nodes;
    dirsage_zero_kernel<<<2048, 256, 0, stream>>>(ws, zero_n);

    const int* src = ei;
    const int* dst = ei + n_edges;
    const long long scatter_threads = (long long)n_edges * 32;
    const int scatter_blocks = (int)((scatter_threads + 255) / 256);
    dirsage_scatter_kernel<<<scatter_blocks, 256, 0, stream>>>(
        x, src, dst, sum_s2d, sum_d2s, cnt_dst, cnt_src, n_edges);

    const int gemm_blocks = (n_nodes + 15) / 16;
    dirsage_wmma_gemm_kernel<<<gemm_blocks, 256, 0, stream>>>(
        x, sum_s2d, sum_d2s, cnt_dst, cnt_src,
        W1, b1, W2, b2, Ws, bs, out, n_nodes);
}